// CenterLoss_81595788689959
// MI455X (gfx1250) — compile-verified
//
#include <hip/hip_runtime.h>
#include <hip/hip_bf16.h>

// ---------------------------------------------------------------------------
// CenterLoss on MI455X (gfx1250, wave32, WMMA).
//
// Dominant work: pd = (n_i + n_j - 2 * C C^T) / 512 over C = centers[labels]
// (4096 x 512 fp32)  ->  17.2 GFLOP fp32 GEMM with a fused masked-reduction
// epilogue (only two scalars survive). C (8 MB) is L2-resident (192 MB L2),
// so this is matrix-core bound: V_WMMA_F32_16X16X4_F32 keeps fp32 precision
// like the reference. Epilogue masking is branchless (v_cndmask, no EXEC
// divergence).
// ---------------------------------------------------------------------------

typedef __attribute__((ext_vector_type(2))) float v2f;
typedef __attribute__((ext_vector_type(8))) float v8f;

#define FEAT 512
#define NB 4096
#define NPAT 50

// ---------------------------------------------------------------------------
// Kernel 0: zero the accumulators + patient histogram in workspace.
// accums[0]=img_sum, accums[1]=intra_sum, accums[2]=inter_sum (doubles)
// ---------------------------------------------------------------------------
__global__ void cl_init_kernel(double* __restrict__ accums, int* __restrict__ hist) {
    int t = threadIdx.x;
    if (t < 3) accums[t] = 0.0;
    if (t < NPAT) hist[t] = 0;
}

// ---------------------------------------------------------------------------
// Kernel 1: gather C = centers[labels], n[b] = ||c_b||^2, image-loss partial
// sum, and histogram of labels_2. One wave (32 lanes) per row; each lane
// handles 4 float4 chunks (32 lanes * 16 floats = 512).
// ---------------------------------------------------------------------------
__global__ __launch_bounds__(256) void cl_gather_kernel(
    const float* __restrict__ x, const int* __restrict__ labels,
    const int* __restrict__ labels2, const float* __restrict__ centers,
    float* __restrict__ Cg, float* __restrict__ nbuf,
    double* __restrict__ accums, int* __restrict__ hist)
{
    const unsigned lane = threadIdx.x & 31u;
    const unsigned wave = threadIdx.x >> 5;     // 8 waves per block
    const unsigned row = blockIdx.x * 8u + wave;

    const float4* xr = (const float4*)(x + (size_t)row * FEAT);
    const int lab = labels[row];
    const float4* cr = (const float4*)(centers + (size_t)lab * FEAT);
    float4* og = (float4*)(Cg + (size_t)row * FEAT);

    float xx = 0.f, cc = 0.f, xc = 0.f;
#pragma unroll
    for (int c = 0; c < 4; ++c) {
        const unsigned idx = lane + 32u * c;
        float4 xv = xr[idx];
        float4 cv = cr[idx];
        og[idx] = cv;
        xx += xv.x * xv.x + xv.y * xv.y + xv.z * xv.z + xv.w * xv.w;
        cc += cv.x * cv.x + cv.y * cv.y + cv.z * cv.z + cv.w * cv.w;
        xc += xv.x * cv.x + xv.y * cv.y + xv.z * cv.z + xv.w * cv.w;
    }
#pragma unroll
    for (int off = 16; off > 0; off >>= 1) {
        xx += __shfl_xor(xx, off, 32);
        cc += __shfl_xor(cc, off, 32);
        xc += __shfl_xor(xc, off, 32);
    }
    if (lane == 0) {
        nbuf[row] = cc;
        float dist = xx + cc - 2.0f * xc;
        dist = fminf(fmaxf(dist, 1e-12f), 1e12f);
        atomicAdd(&accums[0], (double)dist);
        atomicAdd(&hist[labels2[row]], 1);
    }
}

// ---------------------------------------------------------------------------
// Branchless masked accumulate for one pd element.
// ---------------------------------------------------------------------------
__device__ __forceinline__ void cl_accum(float ni, float nj, int li, int lj,
                                         float dot, float inv_d,
                                         float& intra, float& inter)
{
    float pd = fminf(fmaxf((ni + nj - 2.0f * dot) * inv_d, 1e-12f), 1e12f);
    float rec = 1.0f / (1.0f + pd);
    const bool same = (li == lj);
    intra += same ? pd : 0.0f;     // v_cndmask, no EXEC divergence
    inter += same ? 0.0f : rec;
}

// ---------------------------------------------------------------------------
// Kernel 2: fused pairwise GEMM (C * C^T) with masked-sum epilogue.
// Block = 256 threads = 8 waves in a 2 (i) x 4 (j) grid; each wave computes a
// 32x32 macro-tile as 2x2 WMMA 16x16 tiles. Block tile = 64 x 128.
//
// f32 WMMA fragment layouts (ISA 7.12.2):
//   A 16x4:  lane L holds row M=L&15, K = 2*(L>>4) + {0,1} in v[0],v[1]
//   B 4x16:  lane L holds col N=L&15, K = 2*(L>>4) + {0,1}   (C*C^T => same
//            per-lane global_load_b64 pattern as A, no transpose needed)
//   C/D:     element r: M = r + 8*(lane>>4), N = lane&15
// ---------------------------------------------------------------------------
__global__ __launch_bounds__(256) void cl_pair_kernel(
    const float* __restrict__ Cg, const float* __restrict__ nbuf,
    const int* __restrict__ labels2, double* __restrict__ accums)
{
    const unsigned tid = threadIdx.x;
    const unsigned lane = tid & 31u;
    const unsigned wave = tid >> 5;
    const unsigned wi = wave >> 2;              // 0..1
    const unsigned wj = wave & 3u;              // 0..3
    const unsigned i0 = blockIdx.y * 64u + wi * 32u;
    const unsigned j0 = blockIdx.x * 128u + wj * 32u;

    const unsigned m = lane & 15u;
    const unsigned koff = (lane >> 4) * 2u;

    const float* Ar0 = Cg + (size_t)(i0 + m) * FEAT + koff;
    const float* Ar1 = Ar0 + 16 * FEAT;
    const float* Br0 = Cg + (size_t)(j0 + m) * FEAT + koff;
    const float* Br1 = Br0 + 16 * FEAT;

    v8f acc00 = {}, acc01 = {}, acc10 = {}, acc11 = {};

#pragma unroll 4
    for (int k = 0; k < FEAT; k += 4) {
        v2f a0 = *(const v2f*)(Ar0 + k);
        v2f a1 = *(const v2f*)(Ar1 + k);
        v2f b0 = *(const v2f*)(Br0 + k);
        v2f b1 = *(const v2f*)(Br1 + k);
        acc00 = __builtin_amdgcn_wmma_f32_16x16x4_f32(false, a0, false, b0,
                                                      (short)0, acc00, false, false);
        acc01 = __builtin_amdgcn_wmma_f32_16x16x4_f32(false, a0, false, b1,
                                                      (short)0, acc01, false, false);
        acc10 = __builtin_amdgcn_wmma_f32_16x16x4_f32(false, a1, false, b0,
                                                      (short)0, acc10, false, false);
        acc11 = __builtin_amdgcn_wmma_f32_16x16x4_f32(false, a1, false, b1,
                                                      (short)0, acc11, false, false);
    }

    // ---- epilogue: pd -> masked intra/inter partial sums (branchless) ------
    const unsigned hi = lane >> 4;              // 0/1 -> M = r + 8*hi
    const unsigned jl = lane & 15u;
    const float inv_d = 1.0f / (float)FEAT;

    const float nj0 = nbuf[j0 + jl];
    const float nj1 = nbuf[j0 + 16u + jl];
    const int lj0 = labels2[j0 + jl];
    const int lj1 = labels2[j0 + 16u + jl];

    float intra = 0.f, inter = 0.f;
#pragma unroll
    for (int r = 0; r < 8; ++r) {
        const unsigned ia = i0 + (unsigned)r + 8u * hi;
        const unsigned ib = ia + 16u;
        const float ni0 = nbuf[ia];
        const float ni1 = nbuf[ib];
        const int li0 = labels2[ia];
        const int li1 = labels2[ib];

        cl_accum(ni0, nj0, li0, lj0, acc00[r], inv_d, intra, inter);
        cl_accum(ni0, nj1, li0, lj1, acc01[r], inv_d, intra, inter);
        cl_accum(ni1, nj0, li1, lj0, acc10[r], inv_d, intra, inter);
        cl_accum(ni1, nj1, li1, lj1, acc11[r], inv_d, intra, inter);
    }

#pragma unroll
    for (int off = 16; off > 0; off >>= 1) {
        intra += __shfl_xor(intra, off, 32);
        inter += __shfl_xor(inter, off, 32);
    }

    __shared__ float s_intra[8];
    __shared__ float s_inter[8];
    if (lane == 0) { s_intra[wave] = intra; s_inter[wave] = inter; }
    __syncthreads();
    if (tid == 0) {
        double ti = 0.0, te = 0.0;
#pragma unroll
        for (int w = 0; w < 8; ++w) { ti += (double)s_intra[w]; te += (double)s_inter[w]; }
        atomicAdd(&accums[1], ti);
        atomicAdd(&accums[2], te);
    }
}

// ---------------------------------------------------------------------------
// Kernel 3: combine -> scalar output.
// ---------------------------------------------------------------------------
__global__ void cl_final_kernel(const double* __restrict__ accums,
                                const int* __restrict__ hist,
                                const int* __restrict__ y,
                                float* __restrict__ out)
{
    if (threadIdx.x == 0 && blockIdx.x == 0) {
        long long ns = 0;
        for (int p = 0; p < NPAT; ++p) {
            long long h = (long long)hist[p];
            ns += h * h;
        }
        long long nd = (long long)NB * (long long)NB - ns;
        if (ns < 1) ns = 1;
        if (nd < 1) nd = 1;
        double loss = accums[0] / (double)NB;
        if (y[0] != 1) {
            loss += accums[1] / (double)ns + accums[2] / (double)nd;
        }
        out[0] = (float)loss;
    }
}

extern "C" void kernel_launch(void* const* d_in, const int* in_sizes, int n_in,
                              void* d_out, int out_size, void* d_ws, size_t ws_size,
                              hipStream_t stream) {
    (void)in_sizes; (void)n_in; (void)out_size; (void)ws_size;

    const float* x       = (const float*)d_in[0];   // [4096, 512]
    const int*   labels  = (const int*)d_in[1];     // [4096]
    const int*   labels2 = (const int*)d_in[2];     // [4096]
    const int*   y       = (const int*)d_in[3];     // scalar
    const float* centers = (const float*)d_in[4];   // [10000, 512]
    float* out = (float*)d_out;

    char* ws = (char*)d_ws;
    double* accums = (double*)ws;                   // 3 doubles
    int*    hist   = (int*)(ws + 64);               // 50 ints
    float*  nbuf   = (float*)(ws + 512);            // 4096 floats
    float*  Cg     = (float*)(ws + 32768);          // 4096*512 floats (8 MB)

    cl_init_kernel<<<1, 64, 0, stream>>>(accums, hist);

    cl_gather_kernel<<<NB / 8, 256, 0, stream>>>(x, labels, labels2, centers,
                                                 Cg, nbuf, accums, hist);

    dim3 grid(NB / 128, NB / 64);                   // 32 x 64 blocks
    cl_pair_kernel<<<grid, 256, 0, stream>>>(Cg, nbuf, labels2, accums);

    cl_final_kernel<<<1, 1, 0, stream>>>(accums, hist, y, out);
}